// QVAE_CF_72052371358293
// MI455X (gfx1250) — compile-verified
//
#include <hip/hip_runtime.h>
#include <hip/hip_bf16.h>
#include <math.h>

typedef __attribute__((ext_vector_type(2))) float v2f;
typedef __attribute__((ext_vector_type(8))) float v8f;

#define NUM_P 4
#define NUM_C 256
#define CD 16
#define DIM 64
#define LPOS 50
#define LNEG 100
#define LTOT 150

// ---------------------------------------------------------------------------
// Kernel 1: user encoding via argmax over gumbel-perturbed centroid distances.
// Block = 128 threads (4 waves). Wave p handles partition p for 16 users.
// Distance dot-products computed with V_WMMA_F32_16X16X4_F32:
//   A = ue tile   [16 users x 16 dims]   (4 K-steps of 16x4)
//   B = centroids [16 dims x 16 cents]   (per 16-wide N tile, 16 tiles)
//   D = dot tile  [16 users x 16 cents]
// argmax_c( 2*dot - ||c||^2 + gumbel )  (user norm constant over c -> dropped)
// ---------------------------------------------------------------------------
__global__ __launch_bounds__(128) void qvae_encode_user(
    const int*   __restrict__ user_id,     // [B]
    const float* __restrict__ user_emb,    // [NUM_USER, 64]
    const float* __restrict__ centroids,   // [P, C, CD]
    const float* __restrict__ gumbel,      // [B, P, C]
    float*       __restrict__ enc)         // [B, 64]  (workspace)
{
    const int p    = threadIdx.x >> 5;     // wave id == partition
    const int lane = threadIdx.x & 31;
    const int half = lane >> 4;            // 0: lanes 0-15, 1: lanes 16-31
    const int l15  = lane & 15;
    const int b0   = blockIdx.x * 16;

    // ---- A operand: row m = l15 of user sub-vector, lane-half holds K pair.
    // Step kk covers K = 4*kk .. 4*kk+3; this lane holds k = 4*kk + 2*half + {0,1}.
    const int uid = user_id[b0 + l15];
    const float* urow = user_emb + (size_t)uid * DIM + p * CD;
    v2f A[4];
#pragma unroll
    for (int kk = 0; kk < 4; ++kk) {
        const float* s = urow + 4 * kk + 2 * half;
        A[kk][0] = s[0];
        A[kk][1] = s[1];
    }

    float bestv[8];
    int   besti[8];
#pragma unroll
    for (int r = 0; r < 8; ++r) { bestv[r] = -INFINITY; besti[r] = 0; }

    const float* cen_p = centroids + (size_t)p * NUM_C * CD;
    const float* gum_p = gumbel + (size_t)p * NUM_C;

#pragma unroll 1
    for (int tile = 0; tile < NUM_C / 16; ++tile) {
        const int n = tile * 16 + l15;               // this lane's centroid col
        const float* crow = cen_p + (size_t)n * CD;

        // ---- B operand: col n, lane-half holds K pair per step (mirrors A).
        v2f Bm[4];
        float cn = 0.0f;
#pragma unroll
        for (int kk = 0; kk < 4; ++kk) {
            const float* s = crow + 4 * kk + 2 * half;
            Bm[kk][0] = s[0];
            Bm[kk][1] = s[1];
            cn += Bm[kk][0] * Bm[kk][0] + Bm[kk][1] * Bm[kk][1];
        }
        // combine halves -> full ||c_n||^2, lands on lanes with matching l15
        cn += __shfl_xor(cn, 16, 32);

        // ---- 16x16x16 fp32 matmul as 4 chained K=4 WMMAs
        v8f D = {0.f, 0.f, 0.f, 0.f, 0.f, 0.f, 0.f, 0.f};
#pragma unroll
        for (int kk = 0; kk < 4; ++kk) {
            D = __builtin_amdgcn_wmma_f32_16x16x4_f32(
                    false, A[kk], false, Bm[kk], (short)0, D, false, false);
        }

        // ---- score + running argmax; lane owns (rows m = r + 8*half, col n)
#pragma unroll
        for (int r = 0; r < 8; ++r) {
            const int m = r + 8 * half;
            const float g = gum_p[((size_t)(b0 + m)) * (NUM_P * NUM_C) + n];
            const float score = 2.0f * D[r] - cn + g;
            if (score > bestv[r]) { bestv[r] = score; besti[r] = n; }
        }
    }

    // ---- reduce argmax across the 16 lanes of each half (xor 8..1 stays in-half)
#pragma unroll
    for (int r = 0; r < 8; ++r) {
        float v = bestv[r];
        int   i = besti[r];
#pragma unroll
        for (int s = 8; s >= 1; s >>= 1) {
            const float ov = __shfl_xor(v, s, 32);
            const int   oi = __shfl_xor(i, s, 32);
            if (ov > v || (ov == v && oi < i)) { v = ov; i = oi; }
        }
        besti[r] = i;   // now uniform across the half
    }

    // ---- gather winning centroid, write enc[b, p*16 + d], d = l15
#pragma unroll
    for (int r = 0; r < 8; ++r) {
        const int m = r + 8 * half;
        enc[(size_t)(b0 + m) * DIM + p * CD + l15] =
            cen_p[(size_t)besti[r] * CD + l15];
    }
}

// ---------------------------------------------------------------------------
// Kernel 2: item reparameterization + dot with enc. One block per batch row.
// A wave handles one item: lane loads float2 (d = 2*lane, 2*lane+1) -> fully
// coalesced 256B row reads of mu / logvar / eps; shuffle-reduce the dot.
// ---------------------------------------------------------------------------
__global__ __launch_bounds__(256) void qvae_score(
    const float* __restrict__ enc,          // [B, 64]
    const int*   __restrict__ pos_id,       // [B, 50]
    const int*   __restrict__ neg_ids,      // [B, 100]
    const float* __restrict__ item_mu,      // [NI+1, 64]
    const float* __restrict__ item_logvar,  // [NI+1, 64]
    const float* __restrict__ eps_pos,      // [B, 50, 64]
    const float* __restrict__ eps_neg,      // [B, 100, 64]
    float*       __restrict__ out)          // [B, 150]
{
    const int b    = blockIdx.x;
    const int wave = threadIdx.x >> 5;      // 8 waves
    const int lane = threadIdx.x & 31;

    __shared__ float enc_s[DIM];
    if (threadIdx.x < DIM) enc_s[threadIdx.x] = enc[(size_t)b * DIM + threadIdx.x];
    __syncthreads();

    const float e0 = enc_s[2 * lane];
    const float e1 = enc_s[2 * lane + 1];

    for (int l = wave; l < LTOT; l += 8) {
        int id;
        const float* eps;
        if (l < LPOS) {
            id  = pos_id[(size_t)b * LPOS + l];
            eps = eps_pos + ((size_t)b * LPOS + l) * DIM;
        } else {
            const int ln = l - LPOS;
            id  = neg_ids[(size_t)b * LNEG + ln];
            eps = eps_neg + ((size_t)b * LNEG + ln) * DIM;
        }
        const float2 m2 = ((const float2*)(item_mu     + (size_t)id * DIM))[lane];
        const float2 v2 = ((const float2*)(item_logvar + (size_t)id * DIM))[lane];
        const float2 x2 = ((const float2*)eps)[lane];

        const float it0 = m2.x + x2.x * expf(0.5f * v2.x);
        const float it1 = m2.y + x2.y * expf(0.5f * v2.y);
        float acc = e0 * it0 + e1 * it1;
#pragma unroll
        for (int s = 16; s >= 1; s >>= 1) acc += __shfl_xor(acc, s, 32);
        if (lane == 0) out[(size_t)b * LTOT + l] = acc;
    }
}

extern "C" void kernel_launch(void* const* d_in, const int* in_sizes, int n_in,
                              void* d_out, int out_size, void* d_ws, size_t ws_size,
                              hipStream_t stream) {
    const int*   user_id     = (const int*)  d_in[0];
    const int*   pos_id      = (const int*)  d_in[1];
    const int*   neg_ids     = (const int*)  d_in[2];
    const float* user_emb    = (const float*)d_in[3];
    const float* centroids   = (const float*)d_in[4];
    const float* item_mu     = (const float*)d_in[5];
    const float* item_logvar = (const float*)d_in[6];
    const float* eps_pos     = (const float*)d_in[7];
    const float* eps_neg     = (const float*)d_in[8];
    const float* gumbel      = (const float*)d_in[9];
    float* out = (float*)d_out;
    float* enc = (float*)d_ws;               // B * 64 floats = 4 MB scratch

    const int B = in_sizes[0];               // 16384

    qvae_encode_user<<<B / 16, 128, 0, stream>>>(
        user_id, user_emb, centroids, gumbel, enc);
    qvae_score<<<B, 256, 0, stream>>>(
        enc, pos_id, neg_ids, item_mu, item_logvar, eps_pos, eps_neg, out);
}